// GenSP_48095043781092
// MI455X (gfx1250) — compile-verified
//
#include <hip/hip_runtime.h>
#include <cstdint>

typedef float v2f __attribute__((ext_vector_type(2)));
typedef float v4f __attribute__((ext_vector_type(4)));
typedef float v8f __attribute__((ext_vector_type(8)));
typedef unsigned int v4u __attribute__((ext_vector_type(4)));
typedef int v4i __attribute__((ext_vector_type(4)));
typedef int v8i __attribute__((ext_vector_type(8)));

#if defined(__has_builtin)
#if __has_builtin(__builtin_amdgcn_tensor_load_to_lds)
#define USE_TDM 1
#endif
#endif
#ifndef USE_TDM
#define USE_TDM 0
#endif

namespace {
constexpr int B_ = 4, C_ = 64, H_ = 256, W_ = 256;
constexpr int SH = 16, NW_ = 16, NS = 256;
constexpr int P_ = H_ * W_;                              // 65536
constexpr long long OUT_MAIN = (long long)B_ * NS * P_;  // 67108864
}

__device__ __forceinline__ int rel_of(int j) {
  return (j / 3 - 1) * NW_ + (j % 3 - 1);  // {-17,-16,-15,-1,0,1,15,16,17}
}

// ---------------- Kernel A: spix = 16x16 block means -------------------------
__global__ void spix_init_kernel(const float* __restrict__ x,
                                 float* __restrict__ spix) {
  int idx = blockIdx.x * blockDim.x + threadIdx.x;  // (b*C + c)*NS + s
  int s = idx & (NS - 1);
  int bc = idx >> 8;
  int cy = s >> 4, cx = s & 15;
  const float* base = x + (size_t)bc * P_ + (cy * SH) * W_ + cx * SH;
  v4f acc = {};
  for (int r = 0; r < SH; ++r) {
    const v4f* row = (const v4f*)(base + r * W_);  // 64B-aligned
#pragma unroll
    for (int c2 = 0; c2 < 4; ++c2) acc += row[c2];
  }
  spix[idx] = (acc.x + acc.y + acc.z + acc.w) * (1.0f / (SH * SH));
}

// ---------------- Kernels B/D: per-cell WMMA cross + softmax -----------------
// One block per (b, cell). 8 wave32s; GEMM (16x64)x(64x256) via
// v_wmma_f32_16x16x4_f32, K staged in two 32-channel phases (TDM when present).
template <bool FINAL>
__global__ __launch_bounds__(256) void sp_iter_kernel(
    const float* __restrict__ x, const float* __restrict__ spix,
    float* __restrict__ aff_ws, float* __restrict__ out) {
  __shared__ float sX[32 * 256];  // 32 KB: staged pixel channels (half-K)
  __shared__ float sA[16 * 64];   //  4 KB: neighbor spix (M=16, 9 used)
  __shared__ float sC[16 * 256];  // 16 KB: cross[M][q], then aff[9][q]
  __shared__ float sSn[16];

  const int tid = threadIdx.x;
  const int lane = tid & 31;
  const int wave = tid >> 5;
  const int half = lane >> 4;  // 0: K pair {0,1}; 1: K pair {2,3}
  const int lm = lane & 15;

  const int bt = blockIdx.x;
  const int b = bt >> 8;
  const int t = bt & (NS - 1);
  const int cy = t >> 4, cx = t & 15;

  // Stage A: sA[m][c] = spix[b][c][t+rel[m]] (0 if m>=9 or neighbor invalid)
  for (int i = tid; i < 16 * 64; i += 256) {
    int m = i >> 6, c = i & 63;
    float v = 0.f;
    if (m < 9) {
      int sid = t + rel_of(m);
      if (sid >= 0 && sid < NS) v = spix[(b * C_ + c) * NS + sid];
    }
    sA[i] = v;
  }
  __syncthreads();
  if (tid < 16) {
    float sn = 0.f;
    for (int c = 0; c < C_; ++c) { float v = sA[tid * 64 + c]; sn += v * v; }
    sSn[tid] = sn;
  }

  v8f acc0 = {}, acc1 = {};
  float pn = 0.f;
  const size_t xb = (size_t)b * C_ * P_;
  const int pixrow0 = cy * SH * W_ + cx * SH;

  for (int phase = 0; phase < 2; ++phase) {
    __syncthreads();
#if USE_TDM
    // Tensor Data Mover: 3D tile {x=16, y=16, z=32ch} -> packed LDS (sX).
    if (tid < 32) {  // wave 0 issues the DMA (EXEC-independent)
      uint64_t ga =
          (uint64_t)(uintptr_t)(x + xb + (size_t)(phase * 32) * P_ + pixrow0);
      uint32_t la = (uint32_t)(uintptr_t)(&sX[0]);
      v4u g0;
      g0.x = 1u;                 // count=1, user descriptor
      g0.y = la;                 // lds_addr
      g0.z = (uint32_t)ga;       // global_addr[31:0]
      g0.w = (uint32_t)((ga >> 32) & 0x01FFFFFFu) | (2u << 30);  // type=2
      v8i g1;
      g1[0] = (int)(2u << 16);          // data_size=4B; wg_mask=0; no pad/iter
      g1[1] = (int)(256u << 16);        // tensor_dim0 = W (low 16)
      g1[2] = (int)(256u << 16);        // tensor_dim0 hi=0 | tensor_dim1 = H
      g1[3] = (int)(16u << 16);         // tensor_dim1 hi=0 | tile_dim0 = 16
      g1[4] = (int)(16u | (32u << 16)); // tile_dim1 = 16, tile_dim2 = 32
      g1[5] = 256;                      // tensor_dim0_stride = W
      g1[6] = 0;                        // stride0 hi | stride1[15:0] = 0
      g1[7] = 1;                        // tensor_dim1_stride = 65536 (>>16)
      v4i g2 = {64, 0, 0, 0};           // tensor_dim2 = C (OOB bound)
      v4i g3 = {0, 0, 0, 0};
#if __clang_major__ >= 23
      v8i g4 = {};
      __builtin_amdgcn_tensor_load_to_lds(g0, g1, g2, g3, g4, 0);
#else
      __builtin_amdgcn_tensor_load_to_lds(g0, g1, g2, g3, 0);
#endif
      __builtin_amdgcn_s_wait_tensorcnt(0);
    }
#else
    for (int cl = 0; cl < 32; ++cl) {
      int cc = phase * 32 + cl;
      sX[cl * 256 + tid] =
          x[xb + (size_t)cc * P_ + pixrow0 + (tid >> 4) * W_ + (tid & 15)];
    }
#endif
    __syncthreads();
    for (int cl = 0; cl < 32; ++cl) {  // pn partial (thread owns pixel tid)
      float v = sX[cl * 256 + tid];
      pn += v * v;
    }
#pragma unroll
    for (int i = 0; i < 2; ++i) {  // each wave: 2 N-tiles of 16 pixels
      int qbase = (wave * 2 + i) * 16;
      v8f acc = (i == 0) ? acc0 : acc1;
      for (int k0 = 0; k0 < 32; k0 += 4) {
        v2f a, bf;
        int ka = phase * 32 + k0 + half * 2;
        a.x = sA[lm * 64 + ka];
        a.y = sA[lm * 64 + ka + 1];
        int kb = k0 + half * 2;
        bf.x = sX[kb * 256 + qbase + lm];
        bf.y = sX[(kb + 1) * 256 + qbase + lm];
        acc = __builtin_amdgcn_wmma_f32_16x16x4_f32(false, a, false, bf,
                                                    (short)0, acc, false, false);
      }
      if (i == 0) acc0 = acc; else acc1 = acc;
    }
  }
  __syncthreads();
#pragma unroll
  for (int i = 0; i < 2; ++i) {  // D layout: lane n, VGPR r -> M = r + 8*half
    int qcol = (wave * 2 + i) * 16 + lm;
    v8f acc = (i == 0) ? acc0 : acc1;
#pragma unroll
    for (int r = 0; r < 8; ++r) sC[(half * 8 + r) * 256 + qcol] = acc[r];
  }
  __syncthreads();

  // Softmax over 9 neighbors; thread tid owns pixel q = tid
  float dist[9], e[9];
  bool val[9];
  float mind = 3.4e38f;
#pragma unroll
  for (int j = 0; j < 9; ++j) {
    int sid = t + rel_of(j);
    val[j] = (sid >= 0 && sid < NS);
    float d = pn - 2.0f * sC[j * 256 + tid] + sSn[j];
    dist[j] = d;
    if (val[j] && d < mind) mind = d;
  }
  float esum = 0.f;
#pragma unroll
  for (int j = 0; j < 9; ++j) {
    float ev = val[j] ? __expf(mind - dist[j]) : 0.f;
    e[j] = ev;
    esum += ev;
  }
  float inv = 1.0f / esum;
#pragma unroll
  for (int j = 0; j < 9; ++j) e[j] *= inv;

  if (!FINAL) {
#pragma unroll
    for (int j = 0; j < 9; ++j)
      aff_ws[(((size_t)(b * NS + t)) * 9 + j) * 256 + tid] = e[j];
  } else {
    // Park aff in LDS, then one dense vectorized (B128, non-temporal) pass
    // over this cell's output slab: 256 s-rows x 256 pixels.
#pragma unroll
    for (int j = 0; j < 9; ++j) sC[j * 256 + tid] = e[j];
    __syncthreads();
    const int pg = tid & 63;  // 4-pixel group: q0 .. q0+3 (same cell row)
    const int sq = tid >> 6;  // s-quarter
    const int q0 = pg * 4;
    const int pglob = pixrow0 + (q0 >> 4) * W_ + (q0 & 15);
    const size_t ob = (size_t)b * NS * P_ + pglob;
    for (int s = sq * 64; s < sq * 64 + 64; ++s) {
      int d = s - t;  // uniform across the 64 threads of this quarter
      v4f v = {};
#pragma unroll
      for (int j = 0; j < 9; ++j)
        if (d == rel_of(j)) v = *(const v4f*)&sC[j * 256 + q0];
      __builtin_nontemporal_store(v, (v4f*)(out + ob + (size_t)s * P_));
    }
  }
}

// ---------------- Kernel C: spix update (weighted means) ---------------------
__global__ __launch_bounds__(256) void spix_update_kernel(
    const float* __restrict__ x, const float* __restrict__ aff_ws,
    float* __restrict__ spix) {
  __shared__ float sAff[9 * 256];
  __shared__ float red[256];
  __shared__ float part[256];

  const int tid = threadIdx.x;
  const int bs = blockIdx.x;
  const int b = bs >> 8;
  const int s = bs & 255;

  int tcell[9];
#pragma unroll
  for (int j = 0; j < 9; ++j) {
    int tt = s - rel_of(j);  // cell whose slot-j points at s
    tcell[j] = (tt >= 0 && tt < NS) ? tt : -1;
    float v = 0.f;
    if (tcell[j] >= 0)
      v = aff_ws[(((size_t)(b * NS + tcell[j])) * 9 + j) * 256 + tid];
    sAff[j * 256 + tid] = v;
  }
  float rsum = 0.f;
#pragma unroll
  for (int j = 0; j < 9; ++j) rsum += sAff[j * 256 + tid];
  red[tid] = rsum;
  __syncthreads();
  for (int off = 128; off > 0; off >>= 1) {
    if (tid < off) red[tid] += red[tid + off];
    __syncthreads();
  }
  float denom = red[0];

  const int c = tid & 63;
  const int g = tid >> 6;  // 4 pixel-groups of 64
  const size_t xcb = (size_t)(b * C_ + c) * P_;
  float acc = 0.f;
#pragma unroll
  for (int j = 0; j < 9; ++j) {
    if (tcell[j] < 0) continue;
    int tt = tcell[j];
    int base = (tt >> 4) * SH * W_ + (tt & 15) * SH;
    for (int qq = 0; qq < 64; ++qq) {
      int q = g * 64 + qq;
      acc += sAff[j * 256 + q] * x[xcb + base + (q >> 4) * W_ + (q & 15)];
    }
  }
  part[tid] = acc;
  __syncthreads();
  if (tid < 64) {
    float num = part[tid] + part[tid + 64] + part[tid + 128] + part[tid + 192];
    spix[(b * C_ + tid) * NS + s] = num / (denom + 1e-16f);
  }
}

// ---------------- Tail: write ns (and any extra slots) -----------------------
__global__ void tail_kernel(float* __restrict__ out, long long total,
                            long long n) {
  for (long long i = total + threadIdx.x; i < n; i += blockDim.x)
    out[i] = (float)NS;
}

extern "C" void kernel_launch(void* const* d_in, const int* in_sizes, int n_in,
                              void* d_out, int out_size, void* d_ws,
                              size_t ws_size, hipStream_t stream) {
  (void)in_sizes; (void)n_in; (void)ws_size;
  const float* x = (const float*)d_in[0];
  float* spix = (float*)d_ws;                // B*C*NS     = 65536 floats
  float* aff = spix + (size_t)B_ * C_ * NS;  // B*NS*9*256 = 2359296 floats
  float* out = (float*)d_out;

  spix_init_kernel<<<(B_ * C_ * NS) / 256, 256, 0, stream>>>(x, spix);
  sp_iter_kernel<false><<<B_ * NS, 256, 0, stream>>>(x, spix, aff, nullptr);
  spix_update_kernel<<<B_ * NS, 256, 0, stream>>>(x, aff, spix);
  sp_iter_kernel<true><<<B_ * NS, 256, 0, stream>>>(x, spix, nullptr, out);
  if ((long long)out_size > OUT_MAIN)
    tail_kernel<<<1, 64, 0, stream>>>(out, OUT_MAIN, (long long)out_size);
}